// ToposHeytingAttention_40561671143621
// MI455X (gfx1250) — compile-verified
//
#include <hip/hip_runtime.h>
#include <hip/hip_bf16.h>
#include <math.h>

typedef __attribute__((ext_vector_type(2))) float v2f;
typedef __attribute__((ext_vector_type(8))) float v8f;
typedef unsigned int u32x4 __attribute__((ext_vector_type(4)));
typedef int i32x4 __attribute__((ext_vector_type(4)));
typedef int i32x8 __attribute__((ext_vector_type(8)));

#define SEQ 1024
#define DIM 128

__device__ __forceinline__ float wave_max32(float v) {
    #pragma unroll
    for (int o = 16; o > 0; o >>= 1) v = fmaxf(v, __shfl_xor(v, o, 32));
    return v;
}
__device__ __forceinline__ float wave_sum32(float v) {
    #pragma unroll
    for (int o = 16; o > 0; o >>= 1) v += __shfl_xor(v, o, 32);
    return v;
}

// ---------------------------------------------------------------------------
// Kernel 1: Q = sigmoid(x@wq+bq), K = sigmoid(x@wk+bk), V = x@wv+bv
// via V_WMMA_F32_16X16X4_F32 chains.
// ---------------------------------------------------------------------------
__global__ __launch_bounds__(256) void qkv_kernel(
    const float* __restrict__ x,
    const float* __restrict__ wq, const float* __restrict__ bq,
    const float* __restrict__ wk, const float* __restrict__ bk,
    const float* __restrict__ wv, const float* __restrict__ bv,
    float* __restrict__ Qd, float* __restrict__ Kd, float* __restrict__ Vd)
{
    __shared__ float Xs[16][129];

    const int b       = blockIdx.y;
    const int rowbase = blockIdx.x * 16;
    const int tid     = threadIdx.x;
    const int lane    = tid & 31;
    const int wave    = tid >> 5;

    const float* xb = x + (size_t)b * SEQ * DIM + (size_t)rowbase * DIM;
    for (int i = tid; i < 16 * DIM; i += 256)
        Xs[i >> 7][i & 127] = xb[i];
    __syncthreads();

    const int n      = lane & 15;
    const int koff   = (lane >> 4) << 1;
    const int rshift = (lane >> 4) << 3;
    const int dbase  = wave * 16;

    const float* Wm[3] = { wq, wk, wv };
    const float* Bm[3] = { bq, bk, bv };
    float*       Om[3] = { Qd, Kd, Vd };

    for (int m = 0; m < 3; ++m) {
        const float* W = Wm[m];
        v8f acc = {};
        #pragma unroll 4
        for (int k = 0; k < DIM; k += 4) {
            v2f a, bf;
            a[0]  = Xs[n][k + koff];
            a[1]  = Xs[n][k + koff + 1];
            bf[0] = W[(size_t)(k + koff) * DIM + dbase + n];
            bf[1] = W[(size_t)(k + koff + 1) * DIM + dbase + n];
            acc = __builtin_amdgcn_wmma_f32_16x16x4_f32(
                false, a, false, bf, (short)0, acc, false, false);
        }
        const float bias = Bm[m][dbase + n];
        float* O = Om[m] + (size_t)b * SEQ * DIM;
        #pragma unroll
        for (int i = 0; i < 8; ++i) {
            float v = acc[i] + bias;
            if (m < 2) v = 1.0f / (1.0f + expf(-v));
            O[(size_t)(rowbase + i + rshift) * DIM + dbase + n] = v;
        }
    }
}

// ---------------------------------------------------------------------------
// Kernel 2: one WG = (batch, 16-q tile).
// K chunks staged via Tensor Data Mover (tensor_load_to_lds) with HW row
// padding (128 dwords + 2 pad -> 130-float LDS stride, bank-conflict-free).
// Phase 1: implication + online softmax (4 q x 32 k per wave).
// Phase 2: P @ V via V_WMMA_F32_16X16X4_F32.
// ---------------------------------------------------------------------------
__global__ __launch_bounds__(256) void attn_kernel(
    const float* __restrict__ Qd, const float* __restrict__ Kd,
    const float* __restrict__ Vd,
    float* __restrict__ out0, float* __restrict__ at)
{
    __shared__ float Qs[16][128];   // broadcast reads
    __shared__ float Ks[64][130];   // TDM-padded: bank = (2k+d)%64
    __shared__ float Ps[16][65];
    __shared__ float Mp[2][16], Lp[2][16];
    __shared__ float Ms[16], Ls[16];

    const int b     = blockIdx.y;
    const int qbase = blockIdx.x * 16;
    const int tid   = threadIdx.x;
    const int lane  = tid & 31;
    const int wave  = tid >> 5;

    const float* Qb = Qd + (size_t)b * SEQ * DIM + (size_t)qbase * DIM;
    for (int i = tid; i < 16 * DIM; i += 256)
        Qs[i >> 7][i & 127] = Qb[i];

    const int khalf = wave & 1;           // which 32-k half of the chunk
    const int q0l   = (wave >> 1) * 4;    // 4 local q rows per wave
    const int kl    = khalf * 32 + lane;
    float* atb = at + (size_t)b * SEQ * SEQ;
    const float* Kb = Kd + (size_t)b * SEQ * DIM;

    const unsigned int ldsKs = (unsigned int)(size_t)&Ks[0][0];

    // ---- Phase 1: implication scores + online softmax over each k-half ----
    float m[4], l[4];
    #pragma unroll
    for (int j = 0; j < 4; ++j) { m[j] = -3.0e38f; l[j] = 0.0f; }

    for (int c = 0; c < SEQ / 64; ++c) {
        __syncthreads();                  // prior chunk fully consumed
        if (wave == 0) {
            // Tensor DMA: 64x128 f32 tile, row-major, pad 2 dwords per 128.
            const unsigned long long ga =
                (unsigned long long)(size_t)(Kb + (size_t)c * 64 * DIM);
            u32x4 g0;
            g0[0] = 1u;                                     // count=1 (valid)
            g0[1] = ldsKs;                                  // lds_addr
            g0[2] = (unsigned int)(ga & 0xFFFFFFFFu);       // global_addr lo
            g0[3] = (unsigned int)((ga >> 32) & 0x1FFFFFFu) // global_addr hi
                    | 0x80000000u;                          // type=2 (image)
            i32x8 g1;
            g1[0] = (2 << 16)      // data_size = 4B
                  | (1 << 20)      // pad_enable
                  | (6 << 22)      // pad_interval: 128 dwords
                  | (1 << 25);     // pad_amount: 2 dwords
            g1[1] = (DIM & 0xFFFF) << 16;   // tensor_dim0 = 128 (lo16)
            g1[2] = (64  & 0xFFFF) << 16;   // tensor_dim1 = 64  (lo16)
            g1[3] = (DIM & 0xFFFF) << 16;   // tile_dim0 = 128
            g1[4] = 64;                     // tile_dim1 = 64, tile_dim2 = 0
            g1[5] = DIM;                    // tensor_dim0_stride = 128 (lo32)
            g1[6] = 0;
            g1[7] = 0;
            i32x4 gz4 = { 0, 0, 0, 0 };
            i32x8 gz8 = { 0, 0, 0, 0, 0, 0, 0, 0 };
            __builtin_amdgcn_tensor_load_to_lds(g0, g1, gz4, gz4, gz8, 0);
            __builtin_amdgcn_s_wait_tensorcnt(0);
        }
        __syncthreads();                  // Ks ready for all waves

        const int kg = c * 64 + kl;
        float a[4] = { 0.0f, 0.0f, 0.0f, 0.0f };
        #pragma unroll 2
        for (int d = 0; d < DIM; ++d) {
            const float kv = Ks[kl][d];
            #pragma unroll
            for (int j = 0; j < 4; ++j)
                a[j] += (Qs[q0l + j][d] <= kv) ? 1.0f : kv;  // Heyting implication
        }
        #pragma unroll
        for (int j = 0; j < 4; ++j) {
            const int qg = qbase + q0l + j;
            // masked value is 0.0 and STILL enters softmax (reference quirk)
            const float t = (kg > qg) ? 0.0f : a[j] * (1.0f / 128.0f);
            atb[(size_t)qg * SEQ + kg] = t;
            const float s  = 10.0f * t;
            const float nm = fmaxf(m[j], wave_max32(s));
            const float e  = wave_sum32(expf(s - nm));
            l[j] = l[j] * expf(m[j] - nm) + e;
            m[j] = nm;
        }
    }
    // merge the two k-half partials per q (exact log-sum-exp merge)
    if (lane == 0) {
        #pragma unroll
        for (int j = 0; j < 4; ++j) { Mp[khalf][q0l + j] = m[j]; Lp[khalf][q0l + j] = l[j]; }
    }
    __syncthreads();
    if (tid < 16) {
        const float ma = Mp[0][tid], mb = Mp[1][tid];
        const float mm = fmaxf(ma, mb);
        Ms[tid] = mm;
        Ls[tid] = Lp[0][tid] * expf(ma - mm) + Lp[1][tid] * expf(mb - mm);
    }

    // ---- Phase 2: output = softmax @ V via WMMA f32 16x16x4 ----
    const int n      = lane & 15;
    const int koff   = (lane >> 4) << 1;
    const int rshift = (lane >> 4) << 3;
    const int dbase  = wave * 16;
    const float* Vb  = Vd + (size_t)b * SEQ * DIM;

    v8f acc = {};
    for (int c = 0; c < SEQ / 64; ++c) {
        __syncthreads();
        for (int i = tid; i < 16 * 64; i += 256) {
            const int q = i >> 6, kc = i & 63;
            const float tv = atb[(size_t)(qbase + q) * SEQ + c * 64 + kc];
            Ps[q][kc] = expf(10.0f * tv - Ms[q]);
        }
        __syncthreads();

        #pragma unroll 4
        for (int kk = 0; kk < 64; kk += 4) {
            v2f a, bf;
            a[0] = Ps[n][kk + koff];
            a[1] = Ps[n][kk + koff + 1];
            const int kg = c * 64 + kk + koff;
            bf[0] = Vb[(size_t)kg * DIM + dbase + n];
            bf[1] = Vb[(size_t)(kg + 1) * DIM + dbase + n];
            acc = __builtin_amdgcn_wmma_f32_16x16x4_f32(
                false, a, false, bf, (short)0, acc, false, false);
        }
    }

    float* ob = out0 + (size_t)b * SEQ * DIM;
    #pragma unroll
    for (int i = 0; i < 8; ++i) {
        const int row = i + rshift;
        ob[(size_t)(qbase + row) * DIM + dbase + n] = acc[i] * (1.0f / Ls[row]);
    }
}

// ---------------------------------------------------------------------------
extern "C" void kernel_launch(void* const* d_in, const int* in_sizes, int n_in,
                              void* d_out, int out_size, void* d_ws, size_t ws_size,
                              hipStream_t stream) {
    const float* x  = (const float*)d_in[0];
    const float* wq = (const float*)d_in[1];
    const float* bq = (const float*)d_in[2];
    const float* wk = (const float*)d_in[3];
    const float* bk = (const float*)d_in[4];
    const float* wv = (const float*)d_in[5];
    const float* bv = (const float*)d_in[6];

    float* out0 = (float*)d_out;                              // (2,1024,128)
    float* at   = (float*)d_out + (size_t)2 * SEQ * DIM;      // (2,1024,1024)

    float* Qd = (float*)d_ws;
    float* Kd = Qd + (size_t)2 * SEQ * DIM;
    float* Vd = Kd + (size_t)2 * SEQ * DIM;

    dim3 grid(SEQ / 16, 2);
    dim3 block(256);
    qkv_kernel<<<grid, block, 0, stream>>>(x, wq, bq, wk, bk, wv, bv, Qd, Kd, Vd);
    attn_kernel<<<grid, block, 0, stream>>>(Qd, Kd, Vd, out0, at);
}